// RoIBBox_40157944217900
// MI455X (gfx1250) — compile-verified
//
#include <hip/hip_runtime.h>
#include <hip/hip_bf16.h>
#include <math.h>

#define BATCH      64
#define A_TOTAL    8649      // 31*31*9
#define KEEP       6000      // PRE_NMS_TOPN
#define NPAD       6016      // KEEP padded to multiple of 32
#define OUT_N      1500      // POST_NMS_TOPN
#define IOU_THR    0.7f
#define THREADS    1024
#define NWAVES     (THREADS / 32)

#ifndef __has_builtin
#define __has_builtin(x) 0
#endif

#if __has_builtin(__builtin_amdgcn_global_load_async_to_lds_b32)
#define USE_ASYNC_LDS 1
#endif

typedef __attribute__((address_space(1))) int g1_int;   // global (as1) int
typedef __attribute__((address_space(3))) int l3_int;   // LDS (as3) int

__device__ __forceinline__ float neg_inf_f() { return __uint_as_float(0xff800000u); }

__device__ __forceinline__ void wait_async_zero() {
#if __has_builtin(__builtin_amdgcn_s_wait_asynccnt)
    __builtin_amdgcn_s_wait_asynccnt(0);
#else
    asm volatile("s_wait_asynccnt 0" ::: "memory");
#endif
}

// One workgroup per batch image. Entire NMS state lives in LDS (CDNA5: 320KB/WGP).
__global__ __launch_bounds__(THREADS, 1)
void roi_nms_kernel(const float* __restrict__ deltas,   // [B, A, 4] (9-anchor interleave included)
                    const float* __restrict__ probs,    // [B, A]
                    const float* __restrict__ anchors,  // [A, 4] y1,x1,y2,x2
                    float* __restrict__ out)            // [B*OUT_N*4] boxes ++ [B*OUT_N] scores
{
    __shared__ uint32_t sc_bits[A_TOTAL];                 // raw score bits (radix select)
    __shared__ float by1[NPAD], bx1[NPAD], by2[NPAD], bx2[NPAD];
    __shared__ float sc[NPAD];                            // score; -inf == suppressed/inactive
    __shared__ uint32_t hist[256];
    __shared__ float red_val[NWAVES];
    __shared__ int   red_idx[NWAVES];
    __shared__ uint32_t s_prefix, s_mask, s_krem;
    __shared__ uint32_t cnt_gt, cnt_eq;
    __shared__ float bc_box[4];
    __shared__ float bc_val, bc_area;
    __shared__ int   bc_idx;

    const int b   = blockIdx.x;
    const int tid = threadIdx.x;
    const float NEG = neg_inf_f();

    const float* probs_b  = probs  + (size_t)b * A_TOTAL;
    const float* deltas_b = deltas + (size_t)b * A_TOTAL * 4;

    // ---------------- Stage scores into LDS (async global->LDS, ASYNCcnt path) ----
#ifdef USE_ASYNC_LDS
    {
        g1_int* gsrc = (g1_int*)probs_b;     // C-style cast: drops const + addrspace cast
        l3_int* ldst = (l3_int*)sc_bits;
        for (int i = tid; i < A_TOTAL; i += THREADS) {
            __builtin_amdgcn_global_load_async_to_lds_b32(gsrc + i, ldst + i, 0, 0);
        }
        wait_async_zero();
    }
#else
    for (int i = tid; i < A_TOTAL; i += THREADS)
        sc_bits[i] = __float_as_uint(probs_b[i]);
#endif
    if (tid == 0) { s_prefix = 0u; s_mask = 0u; s_krem = KEEP; }
    __syncthreads();

    // ---------------- Exact top-KEEP threshold: 4-pass radix select on float bits --
    // scores are uniform[0,1) => non-negative floats => uint compare == float compare
    for (int shift = 24; shift >= 0; shift -= 8) {
        for (int i = tid; i < 256; i += THREADS) hist[i] = 0u;
        __syncthreads();
        const uint32_t pm = s_mask, pv = s_prefix;
        for (int i = tid; i < A_TOTAL; i += THREADS) {
            uint32_t v = sc_bits[i];
            if ((v & pm) == pv) atomicAdd(&hist[(v >> shift) & 255u], 1u);
        }
        __syncthreads();
        if (tid == 0) {
            uint32_t k = s_krem;
            int bsel = 0;
            for (int bin = 255; bin >= 0; --bin) {
                uint32_t c = hist[bin];
                if (k <= c) { bsel = bin; break; }
                k -= c;
            }
            s_krem   = k;
            s_prefix = pv | ((uint32_t)bsel << shift);
            s_mask   = pm | (0xFFu << shift);
        }
        __syncthreads();
    }
    const uint32_t T        = s_prefix;         // exact KEEP-th largest score bits
    const uint32_t need_eq  = s_krem;           // how many ==T entries we must keep
    const int      gt_count = KEEP - (int)need_eq;

    // ---------------- Compact top-KEEP boxes into LDS, decoding deltas on the fly --
    if (tid == 0) { cnt_gt = 0u; cnt_eq = 0u; }
    __syncthreads();
    for (int i = tid; i < A_TOTAL; i += THREADS) {
        uint32_t v = sc_bits[i];
        int slot = -1;
        if (v > T) {
            slot = (int)atomicAdd(&cnt_gt, 1u);            // exactly gt_count of these
        } else if (v == T) {
            uint32_t e = atomicAdd(&cnt_eq, 1u);
            if (e < need_eq) slot = gt_count + (int)e;
        }
        if (slot >= 0) {
            float a0 = anchors[i * 4 + 0], a1 = anchors[i * 4 + 1];
            float a2 = anchors[i * 4 + 2], a3 = anchors[i * 4 + 3];
            float ah = a2 - a0, aw = a3 - a1;
            float acy = a0 + 0.5f * ah, acx = a1 + 0.5f * aw;
            float d0 = deltas_b[i * 4 + 0] * 0.1f;   // dy
            float d1 = deltas_b[i * 4 + 1] * 0.1f;   // dx
            float d2 = deltas_b[i * 4 + 2] * 0.2f;   // dh
            float d3 = deltas_b[i * 4 + 3] * 0.2f;   // dw
            float h  = expf(d2) * ah;
            float w  = expf(d3) * aw;
            float cy = d0 * ah + acy;
            float cx = d1 * aw + acx;
            by1[slot] = cy - 0.5f * h;
            bx1[slot] = cx - 0.5f * w;
            by2[slot] = cy + 0.5f * h;
            bx2[slot] = cx + 0.5f * w;
            sc [slot] = __uint_as_float(v);
        }
    }
    for (int i = KEEP + tid; i < NPAD; i += THREADS) {     // pad entries inactive
        sc[i] = NEG; by1[i] = bx1[i] = by2[i] = bx2[i] = 0.0f;
    }
    __syncthreads();

    // ---------------- Greedy NMS: 1500 dependent (argmax -> IoU-suppress) steps ----
    float* out_boxes  = out + (size_t)b * OUT_N * 4;
    float* out_scores = out + (size_t)BATCH * OUT_N * 4 + (size_t)b * OUT_N;

    int done_at = OUT_N;
    for (int it = 0; it < OUT_N; ++it) {
        // thread-local masked argmax (suppressed entries are -inf in sc[])
        float bv = NEG; int bi = 0;
        for (int i = tid; i < NPAD; i += THREADS) {
            float v = sc[i];
            if (v > bv) { bv = v; bi = i; }          // strict > keeps lowest index
        }
        // wave32 shuffle reduction (tie -> lower index, matches jnp.argmax)
        for (int m = 16; m >= 1; m >>= 1) {
            float ov = __shfl_xor(bv, m, 32);
            int   oi = __shfl_xor(bi, m, 32);
            if (ov > bv || (ov == bv && oi < bi)) { bv = ov; bi = oi; }
        }
        const int wave = tid >> 5, lane = tid & 31;
        if (lane == 0) { red_val[wave] = bv; red_idx[wave] = bi; }
        __syncthreads();
        if (wave == 0) {
            bv = red_val[lane]; bi = red_idx[lane];
            for (int m = 16; m >= 1; m >>= 1) {
                float ov = __shfl_xor(bv, m, 32);
                int   oi = __shfl_xor(bi, m, 32);
                if (ov > bv || (ov == bv && oi < bi)) { bv = ov; bi = oi; }
            }
            if (lane == 0) {
                bc_val = bv; bc_idx = bi;
                if (bv > NEG) {
                    float y1 = by1[bi], x1 = bx1[bi], y2 = by2[bi], x2 = bx2[bi];
                    bc_box[0] = y1; bc_box[1] = x1; bc_box[2] = y2; bc_box[3] = x2;
                    bc_area   = (y2 - y1) * (x2 - x1);
                    // emit clipped box + score (reference clips boxes to [0,1])
                    out_boxes[(size_t)it * 4 + 0] = fminf(fmaxf(y1, 0.0f), 1.0f);
                    out_boxes[(size_t)it * 4 + 1] = fminf(fmaxf(x1, 0.0f), 1.0f);
                    out_boxes[(size_t)it * 4 + 2] = fminf(fmaxf(y2, 0.0f), 1.0f);
                    out_boxes[(size_t)it * 4 + 3] = fminf(fmaxf(x2, 0.0f), 1.0f);
                    out_scores[it] = bv;
                }
            }
        }
        __syncthreads();
        if (!(bc_val > NEG)) { done_at = it; break; }     // uniform across block

        const float sy1 = bc_box[0], sx1 = bc_box[1];
        const float sy2 = bc_box[2], sx2 = bc_box[3];
        const float sarea = bc_area;
        for (int i = tid; i < NPAD; i += THREADS) {
            float v = sc[i];
            if (v > NEG) {
                float y1 = fmaxf(sy1, by1[i]);
                float x1 = fmaxf(sx1, bx1[i]);
                float y2 = fminf(sy2, by2[i]);
                float x2 = fminf(sx2, bx2[i]);
                float inter = fmaxf(y2 - y1, 0.0f) * fmaxf(x2 - x1, 0.0f);
                float areab = (by2[i] - by1[i]) * (bx2[i] - bx1[i]);
                float iou   = inter / (sarea + areab - inter + 1e-9f);
                if (iou > IOU_THR) sc[i] = NEG;           // suppress (incl. self)
            }
        }
        __syncthreads();
    }

    // zero-fill any remaining output slots (harness poisons d_out)
    for (int k = done_at * 4 + tid; k < OUT_N * 4; k += THREADS) out_boxes[k]  = 0.0f;
    for (int k = done_at + tid;     k < OUT_N;     k += THREADS) out_scores[k] = 0.0f;
}

extern "C" void kernel_launch(void* const* d_in, const int* in_sizes, int n_in,
                              void* d_out, int out_size, void* d_ws, size_t ws_size,
                              hipStream_t stream) {
    (void)in_sizes; (void)n_in; (void)d_ws; (void)ws_size; (void)out_size;
    const float* deltas  = (const float*)d_in[0];   // [64,31,31,36]
    const float* probs   = (const float*)d_in[1];   // [64,31,31,9]
    // d_in[2] = gt_labels (unused by the reference computation)
    const float* anchors = (const float*)d_in[3];   // [8649,4]
    float* out = (float*)d_out;                     // 384000 box floats ++ 96000 scores

    roi_nms_kernel<<<dim3(BATCH), dim3(THREADS), 0, stream>>>(deltas, probs, anchors, out);
}